// HardMaxAttention_68229850464521
// MI455X (gfx1250) — compile-verified
//
#include <hip/hip_runtime.h>
#include <stdint.h>

typedef __attribute__((ext_vector_type(2))) float v2f;
typedef __attribute__((ext_vector_type(8))) float v8f;

#define BB 4
#define TT 4096
#define DM 1024

// ---------------------------------------------------------------------------
// gfx1250 async global->LDS copy (ASYNCcnt-tracked), no VGPR round trip.
// LDS address is taken from the vdst VGPR; generic shared pointer low 32 bits = offset.
// ---------------------------------------------------------------------------
__device__ __forceinline__ void async_g2l_b64(void* lds_ptr, const void* gptr) {
  unsigned lds_off = (unsigned)(uintptr_t)lds_ptr;
  unsigned long long ga = (unsigned long long)(uintptr_t)gptr;
  asm volatile("global_load_async_to_lds_b64 %0, %1, off"
               :: "v"(lds_off), "v"(ga)
               : "memory");
}
__device__ __forceinline__ void wait_async0() {
  asm volatile("s_wait_asynccnt 0x0" ::: "memory");
}

// ---------------------------------------------------------------------------
// Kernel 1: Q/K projections. One wave32 per (b,t) row; 4 dots of length 1024.
// qk[row] = {q0, q1, k0, k1}
// ---------------------------------------------------------------------------
__global__ __launch_bounds__(256) void qk_proj_kernel(
    const float* __restrict__ x, const float* __restrict__ WQ,
    const float* __restrict__ WK, float4* __restrict__ qk) {
  const int wave = threadIdx.x >> 5;
  const int lane = threadIdx.x & 31;
  const int row  = blockIdx.x * 8 + wave;          // row in [0, B*T)
  const float* xr = x + (size_t)row * DM;
  float q0 = 0.f, q1 = 0.f, k0 = 0.f, k1 = 0.f;
  for (int c = lane; c < DM; c += 32) {
    float xv = xr[c];
    q0 = fmaf(xv, WQ[c],      q0);
    q1 = fmaf(xv, WQ[DM + c], q1);
    k0 = fmaf(xv, WK[c],      k0);
    k1 = fmaf(xv, WK[DM + c], k1);
  }
  for (int off = 16; off > 0; off >>= 1) {
    q0 += __shfl_down(q0, off, 32);
    q1 += __shfl_down(q1, off, 32);
    k0 += __shfl_down(k0, off, 32);
    k1 += __shfl_down(k1, off, 32);
  }
  if (lane == 0) qk[row] = make_float4(q0, q1, k0, k1);
}

// ---------------------------------------------------------------------------
// Kernel 2: causal hard-max. Whole K[b] (32 KB) staged in LDS; one thread per
// query t scans s = 0..t, first-occurrence max (strict >) to match jnp.argmax.
// ---------------------------------------------------------------------------
__global__ __launch_bounds__(256) void argmax_kernel(
    const float4* __restrict__ qk, int* __restrict__ idx) {
  __shared__ float2 ks[TT];                         // 32 KB
  const int b = blockIdx.y;
  const float4* qkb = qk + (size_t)b * TT;
  for (int s = threadIdx.x; s < TT; s += 256) {
    float4 v = qkb[s];
    ks[s] = make_float2(v.z, v.w);
  }
  __syncthreads();
  const int t = blockIdx.x * 256 + threadIdx.x;
  float4 v = qkb[t];
  const float q0 = v.x, q1 = v.y;
  float best = fmaf(q0, ks[0].x, q1 * ks[0].y);
  int bi = 0;
  for (int s = 1; s <= t; ++s) {
    float sc = fmaf(q0, ks[s].x, q1 * ks[s].y);
    if (sc > best) { best = sc; bi = s; }
  }
  idx[(size_t)b * TT + t] = bi;
}

// ---------------------------------------------------------------------------
// Kernel 3: gathered-A GEMM  out[g,:] = x[b, idx[g], :] @ W_V^T  via fp32 WMMA.
// 256 threads = 8 waves; M tile = 16 gathered rows, N block = 128 (16/wave).
// K chunked by 64 through DOUBLE-BUFFERED LDS filled with async global->LDS
// loads (ASYNCcnt), so a full chunk of WMMAs overlaps the next chunk's HBM/L2
// traffic. Row stride 66 floats spreads LDS banks for ds_load_b64 fragments.
// Two accumulator chains break the WMMA C->D dependency chain.
// ---------------------------------------------------------------------------
__global__ __launch_bounds__(256) void vgather_gemm_kernel(
    const float* __restrict__ x, const float* __restrict__ WV,
    const int* __restrict__ idx, float* __restrict__ out) {
  __shared__ float ldsA[2][16 * 66];                // 2 x 4.2 KB
  __shared__ float ldsB[2][128 * 66];               // 2 x 33.8 KB

  const int tid = threadIdx.x;
  const int g0  = blockIdx.y * 16;                  // first output row of tile
  const int nb0 = blockIdx.x * 128;                 // first column of block

  // ---- per-thread staging coordinates (constant across K chunks) ----
  const int rA0 = tid >> 5;                         // 0..7
  const int rA1 = rA0 + 8;                          // 8..15
  const int k2  = (tid & 31) << 1;                  // even column 0..62
  const int gA0 = g0 + rA0, gA1 = g0 + rA1;
  const size_t srcA0 = (size_t)((gA0 & ~(TT - 1)) + idx[gA0]) * DM;
  const size_t srcA1 = (size_t)((gA1 & ~(TT - 1)) + idx[gA1]) * DM;

  // ---- fragment coordinates ----
  const int wave = tid >> 5;
  const int lane = tid & 31;
  const int m    = lane & 15;                       // A row / B col in frag
  const int kb   = (lane >> 4) << 1;                // K base: lanes>=16 -> K+2
  const int nloc = wave * 16 + m;                   // local column in ldsB

  v8f acc0 = {}, acc1 = {};
  int cur = 0;

  // ---- issue one K chunk of async global->LDS copies into buffer `buf` ----
  auto issue_chunk = [&](int buf, int kc) {
    // A: 16 gathered rows x 64 cols (2 x b64 per thread)
    async_g2l_b64(&ldsA[buf][rA0 * 66 + k2], x + srcA0 + kc + k2);
    async_g2l_b64(&ldsA[buf][rA1 * 66 + k2], x + srcA1 + kc + k2);
    // B: 128 rows(n) x 64 cols(k) of W_V (16 x b64 per thread, coalesced)
#pragma unroll
    for (int it = 0; it < 16; ++it) {
      int nl = rA0 + it * 8;                        // covers 0..127
      async_g2l_b64(&ldsB[buf][nl * 66 + k2],
                    WV + (size_t)(nb0 + nl) * DM + kc + k2);
    }
    // warm L2 two chunks ahead (global_prefetch_b8)
    if (kc + 64 < DM) {
      int nl = rA0 + (tid & 31) * 4;                // spread over rows
      nl &= 127;
      __builtin_prefetch(WV + (size_t)(nb0 + nl) * DM + kc + 64, 0, 1);
    }
  };

  issue_chunk(0, 0);                                // prologue

  for (int kc = 0; kc < DM; kc += 64) {
    wait_async0();                                  // own copies into cur done
    __syncthreads();                                // everyone's cur ready;
                                                    // everyone done reading nxt
    if (kc + 64 < DM) issue_chunk(cur ^ 1, kc + 64);

    const float* aB = &ldsA[cur][m * 66 + kb];
    const float* bB = &ldsB[cur][nloc * 66 + kb];
#pragma unroll
    for (int kk = 0; kk < 64; kk += 8) {
      v2f a0 = *(const v2f*)(aB + kk);
      v2f b0 = *(const v2f*)(bB + kk);
      acc0 = __builtin_amdgcn_wmma_f32_16x16x4_f32(
          false, a0, false, b0, (short)0, acc0, false, false);
      v2f a1 = *(const v2f*)(aB + kk + 4);
      v2f b1 = *(const v2f*)(bB + kk + 4);
      acc1 = __builtin_amdgcn_wmma_f32_16x16x4_f32(
          false, a1, false, b1, (short)0, acc1, false, false);
    }
    cur ^= 1;
  }

  // D layout: VGPR v, lanes 0-15 -> M=v ; lanes 16-31 -> M=v+8 ; N = lane&15
  const int n   = nb0 + wave * 16 + (lane & 15);
  const int mhi = (lane >> 4) << 3;
#pragma unroll
  for (int v = 0; v < 8; ++v) {
    out[(size_t)(g0 + mhi + v) * DM + n] = acc0[v] + acc1[v];
  }
}

// ---------------------------------------------------------------------------
extern "C" void kernel_launch(void* const* d_in, const int* in_sizes, int n_in,
                              void* d_out, int out_size, void* d_ws,
                              size_t ws_size, hipStream_t stream) {
  (void)in_sizes; (void)n_in; (void)out_size; (void)ws_size;
  const float* x  = (const float*)d_in[0];
  const float* WQ = (const float*)d_in[1];
  const float* WK = (const float*)d_in[2];
  const float* WV = (const float*)d_in[3];
  // d_in[4] = causal mask: causality is applied analytically (scan s<=t).
  float* out = (float*)d_out;

  float4* qk  = (float4*)d_ws;                                   // 256 KB
  int*    idx = (int*)((char*)d_ws + (size_t)BB * TT * sizeof(float4)); // 64 KB

  qk_proj_kernel<<<BB * TT / 8, 256, 0, stream>>>(x, WQ, WK, qk);
  argmax_kernel<<<dim3(TT / 256, BB), 256, 0, stream>>>(qk, idx);
  vgather_gemm_kernel<<<dim3(DM / 128, BB * TT / 16), 256, 0, stream>>>(
      x, WV, idx, out);
}